// L1Wav_54872502174332
// MI455X (gfx1250) — compile-verified
//
#include <hip/hip_runtime.h>

// ---------------------------------------------------------------------------
// L1-wavelet prox (db4, periodized, 4 levels) on MI455X / gfx1250.
// Column-direction filter banks run on the matrix pipe via
// V_WMMA_F32_16X16X4_F32; row-direction passes + thresholding on VALU.
// Filter A-operands come from a compile-time __constant__ table (one
// float4 load per WMMA chunk). All global indices are unsigned 32-bit so
// the compiler emits SGPR-base + 32-bit-VGPR-offset (GVS) addressing
// instead of per-access 64-bit VALU address builds. Global phase provably
// cancels (soft-threshold is phase-equivariant) and is omitted. Roll is
// folded into first/last kernels.
// ---------------------------------------------------------------------------

typedef __attribute__((ext_vector_type(2))) float v2f;
typedef __attribute__((ext_vector_type(8))) float v8f;

constexpr int      Nn     = 4096;
constexpr int      LOGN   = 12;
constexpr unsigned PLu    = 4096u * 4096u;   // plane elements (fits u32)
constexpr size_t   PL     = (size_t)PLu;
constexpr int      LEVELS = 4;
constexpr int      SHIFT  = 1;               // int(round(rng(1000).uniform(-3,3)))
constexpr float    THRESH = 0.005f;          // LAMDA * ALPHA

// db4 analysis filters; hi[k] = (-1)^k * lo[7-k]
constexpr float F_LO[8] = {
    -0.010597401784997278f,  0.032883011666982945f,  0.030841381835986965f,
    -0.18703481171888114f,  -0.02798376941698385f,   0.6308807679295904f,
     0.7148465705525415f,    0.23037781330885523f };
constexpr float F_HI[8] = {
     0.23037781330885523f,  -0.7148465705525415f,    0.6308807679295904f,
     0.02798376941698385f,  -0.18703481171888114f,  -0.030841381835986965f,
     0.032883011666982945f,  0.010597401784997278f };

constexpr float ctap(const float* f, int i) { return (i >= 0 && i < 8) ? f[i] : 0.0f; }

// Compile-time per-lane WMMA A-operand tables.
//  fwd[c][lane] = {lo.x, lo.y, hi.x, hi.y} for analysis chunk c (K = 4c..4c+3)
//  inv[c][lane] = {lo.x, lo.y, hi.x, hi.y} for synthesis chunk c
struct Tabs {
    alignas(16) float fwd[10][32][4];
    alignas(16) float inv[3][32][4];
};
constexpr Tabs make_tabs() {
    Tabs t{};
    for (int c = 0; c < 10; ++c)
        for (int l = 0; l < 32; ++l) {
            const int half = l >> 4, m = l & 15;
            const int kx = 4 * c + 2 * half;     // t index for .x (A[m][K])
            const int ix = kx - 2 * m;           // tap = t - 2m
            t.fwd[c][l][0] = ctap(F_LO, ix);
            t.fwd[c][l][1] = ctap(F_LO, ix + 1);
            t.fwd[c][l][2] = ctap(F_HI, ix);
            t.fwd[c][l][3] = ctap(F_HI, ix + 1);
        }
    for (int c = 0; c < 3; ++c)
        for (int l = 0; l < 32; ++l) {
            const int half = l >> 4, tp = l & 15;
            const int kx = 4 * c + 2 * half;     // m' for .x
            const int ix = tp + 6 - 2 * kx;      // tap = (t0+t') - 2*(mbase+m')
            t.inv[c][l][0] = ctap(F_LO, ix);
            t.inv[c][l][1] = ctap(F_LO, ix - 2);
            t.inv[c][l][2] = ctap(F_HI, ix);
            t.inv[c][l][3] = ctap(F_HI, ix - 2);
        }
    return t;
}
__constant__ Tabs g_tabs = make_tabs();

// -------------------------------- roll in ----------------------------------
__global__ void k_roll_in(const float* __restrict__ xr, const float* __restrict__ xi,
                          float* __restrict__ cre, float* __restrict__ cim) {
    unsigned t = blockIdx.x * blockDim.x + threadIdx.x;
    unsigned i = t >> LOGN, j = t & (Nn - 1);
    unsigned s = ((i - SHIFT) & (Nn - 1)) * Nn + ((j - SHIFT) & (Nn - 1));
    cre[t] = xr[s];
    cim[t] = xi[s];
}

// ------------------- vertical analysis (WMMA f32 16x16x4) ------------------
// Per wave: one 16-col x 16-row tile of both approx (a) and detail (d).
// O = sum_{c=0..9} F_c(16x4) * X_c(4x16),  F[m][t] = filt[t - 2m].
__global__ void __launch_bounds__(128)
k_dwt_v(const float* __restrict__ Cbase, float* __restrict__ Tbase, int n) {
    const unsigned n2 = (unsigned)n >> 1;
    const float* src = Cbase + (size_t)blockIdx.z * PL;
    float*       dst = Tbase + (size_t)blockIdx.z * PL;
    const unsigned lane = threadIdx.x;
    const unsigned half = lane >> 4;     // A/B K-pair select (K = 2*half, 2*half+1)
    const unsigned l15  = lane & 15;     // A: output row m ; B/D: column
    const unsigned j    = (blockIdx.x * 4 + threadIdx.y) * 16 + l15;
    const unsigned m0   = blockIdx.y * 16;
    const unsigned rowbase = 2 * m0;
    const unsigned nm1  = (unsigned)n - 1;

    v8f accA = {};
    v8f accD = {};
#pragma unroll
    for (int c = 0; c < 10; ++c) {
        const float4 f = *(const float4*)(&g_tabs.fwd[c][lane][0]);
        v2f alo, ahi, b;
        alo.x = f.x;  alo.y = f.y;
        ahi.x = f.z;  ahi.y = f.w;
        const unsigned kx = 4u * c + 2u * half;
        const unsigned r0 = (rowbase + kx)      & nm1;
        const unsigned r1 = (rowbase + kx + 1u) & nm1;
        b.x = src[r0 * (unsigned)Nn + j];
        b.y = src[r1 * (unsigned)Nn + j];
        accA = __builtin_amdgcn_wmma_f32_16x16x4_f32(false, alo, false, b,
                                                     (short)0, accA, false, false);
        accD = __builtin_amdgcn_wmma_f32_16x16x4_f32(false, ahi, false, b,
                                                     (short)0, accD, false, false);
    }
#pragma unroll
    for (int r = 0; r < 8; ++r) {            // D: VGPR r -> M = r + 8*half
        const unsigned mo = m0 + (unsigned)r + 8u * half;
        dst[mo        * (unsigned)Nn + j] = accA[r];
        dst[(n2 + mo) * (unsigned)Nn + j] = accD[r];
    }
}

// ------------------------ horizontal analysis (VALU) -----------------------
__global__ void k_dwt_h(const float* __restrict__ Tbase, float* __restrict__ Cbase, int n) {
    const unsigned n2 = (unsigned)n >> 1;
    const float* src = Tbase + (size_t)blockIdx.z * PL;
    float*       dst = Cbase + (size_t)blockIdx.z * PL;
    const unsigned i = blockIdx.y;
    const unsigned m = blockIdx.x * blockDim.x + threadIdx.x;
    const unsigned nm1 = (unsigned)n - 1;
    const float* row = src + i * (unsigned)Nn;
    float a = 0.0f, d = 0.0f;
#pragma unroll
    for (int k = 0; k < 8; ++k) {
        const float x = row[(2u * m + (unsigned)k) & nm1];
        a += F_LO[k] * x;
        d += F_HI[k] * x;
    }
    dst[i * (unsigned)Nn + m]      = a;
    dst[i * (unsigned)Nn + n2 + m] = d;
}

// ------------------- complex soft-threshold (whole pyramid) ----------------
__global__ void k_soft(float* __restrict__ cre, float* __restrict__ cim) {
    unsigned t = blockIdx.x * blockDim.x + threadIdx.x;
    const float re = cre[t], im = cim[t];
    const float mag = sqrtf(re * re + im * im);
    const float scale = (mag > THRESH) ? (1.0f - THRESH / mag) : 0.0f;
    cre[t] = re * scale;
    cim[t] = im * scale;
}

// ----------------------- horizontal synthesis (VALU) -----------------------
__global__ void k_idwt_h(const float* __restrict__ Cbase, float* __restrict__ Tbase, int n) {
    const unsigned n2 = (unsigned)n >> 1;
    const float* src = Cbase + (size_t)blockIdx.z * PL;
    float*       dst = Tbase + (size_t)blockIdx.z * PL;
    const unsigned i = blockIdx.y;
    const unsigned t = blockIdx.x * blockDim.x + threadIdx.x;
    const unsigned nm1 = (unsigned)n - 1;
    const float* row = src + i * (unsigned)Nn;
    float x = 0.0f;
#pragma unroll
    for (int k = 0; k < 8; ++k) {
        const unsigned idx = (t - (unsigned)k) & nm1;
        if ((idx & 1u) == 0u) {
            const unsigned h = idx >> 1;
            x += F_LO[k] * row[h] + F_HI[k] * row[n2 + h];
        }
    }
    dst[i * (unsigned)Nn + t] = x;
}

// -------------------- vertical synthesis (WMMA f32 16x16x4) ----------------
// X[t][j] = sum_m a[m][j]*lo[t-2m] + d[m][j]*hi[t-2m]; a 16-row tile needs
// m in [t0/2-3, t0/2+7] -> 3 chunks of 4 for a and for d (6 WMMAs / tile).
__global__ void __launch_bounds__(128)
k_idwt_v(const float* __restrict__ Tbase, float* __restrict__ Cbase, int n) {
    const unsigned n2 = (unsigned)n >> 1;
    const float* src = Tbase + (size_t)blockIdx.z * PL;
    float*       dst = Cbase + (size_t)blockIdx.z * PL;
    const unsigned lane = threadIdx.x;
    const unsigned half = lane >> 4;
    const unsigned l15  = lane & 15;
    const unsigned j    = (blockIdx.x * 4 + threadIdx.y) * 16 + l15;
    const unsigned t0   = blockIdx.y * 16;
    const unsigned mbase = (t0 >> 1) + n2 - 3u;   // pre-biased by +n2; t0-2*(mbase-n2)==6

    v8f acc = {};
#pragma unroll
    for (int c = 0; c < 3; ++c) {
        const float4 f = *(const float4*)(&g_tabs.inv[c][lane][0]);
        v2f flo, fhi, bA, bD;
        flo.x = f.x;  flo.y = f.y;
        fhi.x = f.z;  fhi.y = f.w;
        const unsigned kx  = 4u * c + 2u * half;
        const unsigned mA0 = (mbase + kx)      & (n2 - 1u);
        const unsigned mA1 = (mbase + kx + 1u) & (n2 - 1u);
        bA.x = src[mA0        * (unsigned)Nn + j];
        bA.y = src[mA1        * (unsigned)Nn + j];
        bD.x = src[(n2 + mA0) * (unsigned)Nn + j];
        bD.y = src[(n2 + mA1) * (unsigned)Nn + j];
        acc = __builtin_amdgcn_wmma_f32_16x16x4_f32(false, flo, false, bA,
                                                    (short)0, acc, false, false);
        acc = __builtin_amdgcn_wmma_f32_16x16x4_f32(false, fhi, false, bD,
                                                    (short)0, acc, false, false);
    }
#pragma unroll
    for (int r = 0; r < 8; ++r)
        dst[(t0 + (unsigned)r + 8u * half) * (unsigned)Nn + j] = acc[r];
}

// -------------------------------- roll out ---------------------------------
__global__ void k_roll_out(const float* __restrict__ cre, const float* __restrict__ cim,
                           float2* __restrict__ out) {
    unsigned t = blockIdx.x * blockDim.x + threadIdx.x;
    unsigned i = t >> LOGN, j = t & (Nn - 1);
    unsigned s = ((i + SHIFT) & (Nn - 1)) * Nn + ((j + SHIFT) & (Nn - 1));
    out[t] = make_float2(cre[s], cim[s]);   // interleaved complex64
}

// ---------------------------------------------------------------------------
extern "C" void kernel_launch(void* const* d_in, const int* in_sizes, int n_in,
                              void* d_out, int out_size, void* d_ws, size_t ws_size,
                              hipStream_t stream) {
    (void)in_sizes; (void)n_in; (void)out_size; (void)ws_size;
    const float* xr = (const float*)d_in[0];
    const float* xi = (const float*)d_in[1];
    float* C = (float*)d_ws;          // coefficient planes: re, im (contiguous)
    float* T = C + 2 * PL;            // temp planes: re, im (contiguous)
    float2* out = (float2*)d_out;

    k_roll_in<<<(unsigned)(PL / 256), 256, 0, stream>>>(xr, xi, C, C + PL);

    for (int lev = 0; lev < LEVELS; ++lev) {
        const int n = Nn >> lev;
        dim3 gv(n / 64, n / 32, 2);                 // col-tiles/4 x row-tiles x planes
        k_dwt_v<<<gv, dim3(32, 4), 0, stream>>>(C, T, n);
        dim3 gh((n / 2) / 256 ? (n / 2) / 256 : 1, n, 2);
        k_dwt_h<<<gh, 256, 0, stream>>>(T, C, n);
    }

    k_soft<<<(unsigned)(PL / 256), 256, 0, stream>>>(C, C + PL);

    for (int lev = LEVELS - 1; lev >= 0; --lev) {
        const int n = Nn >> lev;
        dim3 gh(n / 256, n, 2);
        k_idwt_h<<<gh, 256, 0, stream>>>(C, T, n);
        dim3 gs(n / 64, n / 16, 2);
        k_idwt_v<<<gs, dim3(32, 4), 0, stream>>>(T, C, n);
    }

    k_roll_out<<<(unsigned)(PL / 256), 256, 0, stream>>>(C, C + PL, out);
}